// Resnet50RoIHead_14731737825436
// MI455X (gfx1250) — compile-verified
//
#include <hip/hip_runtime.h>

// ---------------------------------------------------------------------------
// Problem constants (fixed by the reference's setup_inputs)
// ---------------------------------------------------------------------------
constexpr int Nimg  = 2;
constexpr int Cch   = 1024;   // channels
constexpr int Hf    = 25;
constexpr int Wfm   = 25;
constexpr int Rroi  = 128;
constexpr int Pp    = 7;      // roi bins per dim
constexpr int PP    = 49;     // Pp*Pp
constexpr int Dd    = 2048;   // sketch dim / fc7 dim
constexpr int NCOLS = Rroi * PP;  // 6272 = 98*64

typedef float v2f __attribute__((ext_vector_type(2)));
typedef float v8f __attribute__((ext_vector_type(8)));

// GEMM tiling
#define BM 64
#define BN 128
#define BK 32
#define AKP 36    // A LDS row stride: 36*4B = 144B (16B-aligned rows, even)
#define BNP 132   // B LDS row stride: 132*4B = 528B (16B-aligned rows)

// ---------------------------------------------------------------------------
// CDNA5 async global->LDS staging (ASYNCcnt path), with guarded fallback.
// Builtin signature (per hipcc diagnostic): (v4i addrspace(1)*, v4i addrspace(3)*,
// imm offset, imm cpol) where v4i = int __attribute__((vector_size(16))).
// ---------------------------------------------------------------------------
#if defined(__has_builtin)
#if __has_builtin(__builtin_amdgcn_global_load_async_to_lds_b128) && \
    __has_builtin(__builtin_amdgcn_s_wait_asynccnt)
#define USE_ASYNC_LDS 1
#endif
#endif

#if defined(USE_ASYNC_LDS)
typedef int v4i __attribute__((vector_size(16)));
typedef __attribute__((address_space(1))) v4i g_v4i;
typedef __attribute__((address_space(3))) v4i l_v4i;
// addrspace(3) pointers are 32-bit; low 32 bits of a flat LDS address are the
// LDS byte offset (ISA aperture rule), so truncation is the correct cast.
__device__ __forceinline__ l_v4i* to_lds(void* p) {
  return (l_v4i*)(unsigned int)(unsigned long long)p;
}
__device__ __forceinline__ g_v4i* to_glob(const void* p) {
  return (g_v4i*)(unsigned long long)p;
}
#endif

// ---------------------------------------------------------------------------
// RoI max pool (both streams). Writes PA/PB as [col=r*49+bin][c] row-major,
// and the 49-bin mean (stream 0 only) for the fc7 GEMM.
// ---------------------------------------------------------------------------
__global__ __launch_bounds__(256)
void roi_pool_kernel(const float* __restrict__ x, const float* __restrict__ xn,
                     const float* __restrict__ rois, const int* __restrict__ ridx,
                     const int* __restrict__ img_h, const int* __restrict__ img_w,
                     float* __restrict__ PA, float* __restrict__ PB,
                     float* __restrict__ feat) {
  const int r = blockIdx.x;
  const int stream = blockIdx.y;
  const float sx = (float)Wfm / (float)img_w[0];
  const float sy = (float)Hf  / (float)img_h[0];
  // jnp.round == round-half-to-even == rintf (RNE default)
  const float x1 = rintf(rois[r * 4 + 0] * sx);
  const float y1 = rintf(rois[r * 4 + 1] * sy);
  const float x2 = rintf(rois[r * 4 + 2] * sx);
  const float y2 = rintf(rois[r * 4 + 3] * sy);
  const float bw = fmaxf(x2 - x1 + 1.0f, 1.0f) / (float)Pp;
  const float bh = fmaxf(y2 - y1 + 1.0f, 1.0f) / (float)Pp;

  int wsA[Pp], weA[Pp], hsA[Pp], heA[Pp];
#pragma unroll
  for (int p = 0; p < Pp; ++p) {
    wsA[p] = (int)fminf(fmaxf(floorf((float)p * bw) + x1, 0.0f), (float)Wfm);
    weA[p] = (int)fminf(fmaxf(ceilf((float)(p + 1) * bw) + x1, 0.0f), (float)Wfm);
    hsA[p] = (int)fminf(fmaxf(floorf((float)p * bh) + y1, 0.0f), (float)Hf);
    heA[p] = (int)fminf(fmaxf(ceilf((float)(p + 1) * bh) + y1, 0.0f), (float)Hf);
  }

  const float* __restrict__ src = (stream == 0) ? x : xn;
  float* __restrict__ dst = (stream == 0) ? PA : PB;
  const int b = ridx[r];
  const float* __restrict__ f0 = src + (size_t)b * Cch * Hf * Wfm;

  for (int c = threadIdx.x; c < Cch; c += blockDim.x) {
    const float* __restrict__ fc = f0 + (size_t)c * Hf * Wfm;
    float accum = 0.0f;
#pragma unroll
    for (int ph = 0; ph < Pp; ++ph) {
#pragma unroll
      for (int pw = 0; pw < Pp; ++pw) {
        float v = -__builtin_inff();
        for (int yy = hsA[ph]; yy < heA[ph]; ++yy)
          for (int xx = wsA[pw]; xx < weA[pw]; ++xx)
            v = fmaxf(v, fc[yy * Wfm + xx]);
        v = __builtin_isfinite(v) ? v : 0.0f;   // empty bin -> 0
        const int col = r * PP + ph * Pp + pw;
        dst[(size_t)col * Cch + c] = v;
        accum += v;
      }
    }
    if (stream == 0) feat[(size_t)r * Cch + c] = accum * (1.0f / (float)PP);
  }
}

// ---------------------------------------------------------------------------
// Bilinear-form matrix: M[c2][m*1024+c1] = s1[c1]*s2[c2]*Ws[(h1[c1]+h2[c2])&2047][m]
// (collapses the count-sketch circular convolution projected onto Ws)
// ---------------------------------------------------------------------------
__global__ __launch_bounds__(256)
void make_m_kernel(const float* __restrict__ Ws, const float* __restrict__ s1,
                   const float* __restrict__ s2, const int* __restrict__ h1,
                   const int* __restrict__ h2, float* __restrict__ M) {
  const int idx = blockIdx.x * blockDim.x + threadIdx.x;
  if (idx >= Cch * Dd) return;
  const int n  = idx & (Dd - 1);
  const int c2 = idx / Dd;
  const int m  = n >> 10;
  const int c1 = n & 1023;
  const int k  = (h1[c1] + h2[c2]) & (Dd - 1);
  M[idx] = s1[c1] * s2[c2] * Ws[k * 2 + m];
}

// ---------------------------------------------------------------------------
// FP32 WMMA GEMM:  C[M x 2048] = A[M x 1024] * B[1024 x 2048]
// 256 threads = 8 waves; block tile 64x128, wave tile 32x32 (2x2 WMMA tiles).
// Tiles staged to LDS via GLOBAL_LOAD_ASYNC_TO_LDS_B128 (ASYNCcnt) when the
// toolchain exposes the builtins; otherwise register-staged float4 copies.
// A LDS layout [m][k] (b64 fragments); B LDS layout [k][n] (b32-pair frags).
// fuseScore==1: instead of storing C, multiply elementwise with PA and
// emit deterministic per-block roi partial sums.
// ---------------------------------------------------------------------------
__global__ __launch_bounds__(256)
void wmma_gemm_kernel(const float* __restrict__ A, const float* __restrict__ B,
                      float* __restrict__ Cout, int Mrows, int fuseScore,
                      const float* __restrict__ PA, float* __restrict__ scoreP) {
  __shared__ float Als[BM][AKP];
  __shared__ float Bls[BK][BNP];   // row-major [k][n]
  __shared__ float rbuf[256];

  const int tid  = threadIdx.x;
  const int lane = tid & 31;
  const int wave = tid >> 5;
  const int wm   = wave & 1;   // wave row  (2)
  const int wn   = wave >> 1;  // wave col  (4)
  const int row0 = blockIdx.y * BM;
  const int col0 = blockIdx.x * BN;

  const int lrow  = lane & 15;           // m/n within 16x16 tile
  const int khalf = (lane >> 4) << 1;    // 0 or 2 (A/B fragment k-pair)

  const v8f vzero = {0.f, 0.f, 0.f, 0.f, 0.f, 0.f, 0.f, 0.f};
  v8f acc[2][2];
  acc[0][0] = vzero; acc[0][1] = vzero; acc[1][0] = vzero; acc[1][1] = vzero;

  const int ar_ld = tid >> 3;          // 0..31 (A tile rows, 2 passes)
  const int ak_ld = (tid & 7) * 4;     // k within tile, float4 chunk
  const int bk_ld = tid >> 5;          // 0..7 (B tile k rows, 4 passes)
  const int bn_ld = (tid & 31) * 4;    // n within tile, float4 chunk

  for (int k0 = 0; k0 < Cch; k0 += BK) {
#if defined(USE_ASYNC_LDS)
    // --- async stage A tile 64x32: 2 x b128 per thread ---
#pragma unroll
    for (int i = 0; i < 2; ++i) {
      const int rr = ar_ld + i * 32;
      __builtin_amdgcn_global_load_async_to_lds_b128(
          to_glob(A + (size_t)(row0 + rr) * Cch + k0 + ak_ld),
          to_lds(&Als[rr][ak_ld]), 0, 0);
    }
    // --- async stage B tile 32x128: 4 x b128 per thread ---
#pragma unroll
    for (int i = 0; i < 4; ++i) {
      const int kk = bk_ld + i * 8;
      __builtin_amdgcn_global_load_async_to_lds_b128(
          to_glob(B + (size_t)(k0 + kk) * Dd + col0 + bn_ld),
          to_lds(&Bls[kk][bn_ld]), 0, 0);
    }
    __builtin_amdgcn_s_wait_asynccnt(0);
#else
    // --- fallback: register-staged copies ---
#pragma unroll
    for (int i = 0; i < 2; ++i) {
      const int rr = ar_ld + i * 32;
      const float4 va = *(const float4*)(A + (size_t)(row0 + rr) * Cch + k0 + ak_ld);
      *(float4*)&Als[rr][ak_ld] = va;
    }
#pragma unroll
    for (int i = 0; i < 4; ++i) {
      const int kk = bk_ld + i * 8;
      const float4 vb = *(const float4*)(B + (size_t)(k0 + kk) * Dd + col0 + bn_ld);
      *(float4*)&Bls[kk][bn_ld] = vb;
    }
    if (k0 + BK < Cch) {  // prefetch next K tile (global_prefetch_b8)
      __builtin_prefetch(A + (size_t)(row0 + ar_ld) * Cch + k0 + BK + ak_ld, 0, 3);
      __builtin_prefetch(B + (size_t)(k0 + BK + bk_ld) * Dd + col0 + bn_ld, 0, 3);
    }
#endif
    __syncthreads();

#pragma unroll
    for (int kk = 0; kk < BK; kk += 4) {
      v2f afrag[2], bfrag[2];
#pragma unroll
      for (int t = 0; t < 2; ++t) {
        afrag[t] = *(const v2f*)&Als[wm * 32 + t * 16 + lrow][kk + khalf];
        const int bn = wn * 32 + t * 16 + lrow;
        bfrag[t].x = Bls[kk + khalf + 0][bn];
        bfrag[t].y = Bls[kk + khalf + 1][bn];
      }
#pragma unroll
      for (int tm = 0; tm < 2; ++tm)
#pragma unroll
        for (int tn = 0; tn < 2; ++tn)
          acc[tm][tn] = __builtin_amdgcn_wmma_f32_16x16x4_f32(
              false, afrag[tm], false, bfrag[tn], (short)0, acc[tm][tn],
              false, false);
    }
    __syncthreads();
  }

  if (!fuseScore) {
    // plain store: C/D layout -> row = v + 8*(lane>=16), col = lane%16
#pragma unroll
    for (int tm = 0; tm < 2; ++tm)
#pragma unroll
      for (int tn = 0; tn < 2; ++tn) {
        const int n = col0 + wn * 32 + tn * 16 + lrow;
#pragma unroll
        for (int v = 0; v < 8; ++v) {
          const int row = row0 + wm * 32 + tm * 16 + v + ((lane >> 4) << 3);
          if (row < Mrows) Cout[(size_t)row * Dd + n] = acc[tm][tn][v];
        }
      }
  } else {
    // fused: p = T[col][n] * PA[col][c1]; bucket by roi (<=3 rois per 64 cols)
    const int rbase = row0 / PP;
    float part[4] = {0.f, 0.f, 0.f, 0.f};
#pragma unroll
    for (int tm = 0; tm < 2; ++tm)
#pragma unroll
      for (int tn = 0; tn < 2; ++tn) {
        const int n  = col0 + wn * 32 + tn * 16 + lrow;
        const int c1 = n & 1023;
#pragma unroll
        for (int v = 0; v < 8; ++v) {
          const int row = row0 + wm * 32 + tm * 16 + v + ((lane >> 4) << 3);
          const float p = acc[tm][tn][v] * PA[(size_t)row * Cch + c1];
          const int dr = row / PP - rbase;
          part[0] += (dr == 0) ? p : 0.f;
          part[1] += (dr == 1) ? p : 0.f;
          part[2] += (dr == 2) ? p : 0.f;
          part[3] += (dr == 3) ? p : 0.f;
        }
      }
    // deterministic block tree-reduction per roi slot
    for (int k = 0; k < 4; ++k) {
      rbuf[tid] = part[k];
      __syncthreads();
      for (int s = 128; s > 0; s >>= 1) {
        if (tid < s) rbuf[tid] += rbuf[tid + s];
        __syncthreads();
      }
      if (tid == 0)
        scoreP[((size_t)blockIdx.y * gridDim.x + blockIdx.x) * 4 + k] = rbuf[0];
      __syncthreads();
    }
  }
}

// ---------------------------------------------------------------------------
// roi_bbox[r][j] = bb[j] + sum_n (fc7[r][n] + bc[n]) * Wb[n][j]
// ---------------------------------------------------------------------------
__global__ __launch_bounds__(256)
void bbox_kernel(const float* __restrict__ fc7, const float* __restrict__ bc,
                 const float* __restrict__ Wb, const float* __restrict__ bb,
                 float* __restrict__ out) {
  __shared__ float sh[256];
  const int r = blockIdx.x;
  float pj[4] = {0.f, 0.f, 0.f, 0.f};
  for (int n = threadIdx.x; n < Dd; n += 256) {
    const float f = fc7[(size_t)r * Dd + n] + bc[n];
    pj[0] += f * Wb[n * 4 + 0];
    pj[1] += f * Wb[n * 4 + 1];
    pj[2] += f * Wb[n * 4 + 2];
    pj[3] += f * Wb[n * 4 + 3];
  }
#pragma unroll
  for (int j = 0; j < 4; ++j) {
    sh[threadIdx.x] = pj[j];
    __syncthreads();
    for (int s = 128; s > 0; s >>= 1) {
      if (threadIdx.x < s) sh[threadIdx.x] += sh[threadIdx.x + s];
      __syncthreads();
    }
    if (threadIdx.x == 0) out[r * 4 + j] = sh[0] + bb[j];
    __syncthreads();
  }
}

// ---------------------------------------------------------------------------
// Deterministic final reduce of GEMM2 per-block partials into roi_scores.
// block (bx, by): m = bx>=8, rows [by*64, by*64+63], slot k -> r = by*64/49 + k
// ---------------------------------------------------------------------------
__global__ void score_final_kernel(const float* __restrict__ scoreP,
                                   const float* __restrict__ bs,
                                   float* __restrict__ outScores) {
  const int t = threadIdx.x;
  if (t >= Rroi * 2) return;
  const int r = t >> 1;
  const int m = t & 1;
  float sum = 0.f;
  for (int by = 0; by < NCOLS / BM; ++by) {
    const int k = r - (by * BM) / PP;
    if (k < 0 || k > 3) continue;
    for (int bx = m * 8; bx < m * 8 + 8; ++bx)
      sum += scoreP[(by * 16 + bx) * 4 + k];
  }
  outScores[r * 2 + m] = bs[m] + sum;
}

// ---------------------------------------------------------------------------
extern "C" void kernel_launch(void* const* d_in, const int* in_sizes, int n_in,
                              void* d_out, int out_size, void* d_ws, size_t ws_size,
                              hipStream_t stream) {
  const float* x    = (const float*)d_in[0];
  const float* xn   = (const float*)d_in[1];
  const float* rois = (const float*)d_in[2];
  const int*   ridx = (const int*)d_in[3];
  const float* Wc   = (const float*)d_in[4];
  const float* bc   = (const float*)d_in[5];
  const float* Wb   = (const float*)d_in[6];
  const float* bb   = (const float*)d_in[7];
  const float* Ws   = (const float*)d_in[8];
  const float* bs   = (const float*)d_in[9];
  const float* s1   = (const float*)d_in[10];
  const float* s2   = (const float*)d_in[11];
  const int*   h1   = (const int*)d_in[12];
  const int*   h2   = (const int*)d_in[13];
  const int* img_h  = (const int*)d_in[14];
  const int* img_w  = (const int*)d_in[15];
  (void)in_sizes; (void)n_in; (void)out_size; (void)ws_size;

  float* ws     = (float*)d_ws;
  float* PA     = ws;                                  // [6272][1024]
  float* PB     = PA + (size_t)NCOLS * Cch;            // [6272][1024]
  float* feat   = PB + (size_t)NCOLS * Cch;            // [128][1024]
  float* Mb     = feat + (size_t)Rroi * Cch;           // [1024][2048]
  float* fc7    = Mb + (size_t)Cch * Dd;               // [128][2048]
  float* scoreP = fc7 + (size_t)Rroi * Dd;             // [98*16][4]
  float* out    = (float*)d_out;                       // bbox[512] ++ scores[256]

  roi_pool_kernel<<<dim3(Rroi, 2), 256, 0, stream>>>(x, xn, rois, ridx, img_h,
                                                     img_w, PA, PB, feat);
  make_m_kernel<<<(Cch * Dd) / 256, 256, 0, stream>>>(Ws, s1, s2, h1, h2, Mb);
  // GEMM1: fc7 = feat @ Wc   (128 x 1024 x 2048)
  wmma_gemm_kernel<<<dim3(Dd / BN, Rroi / BM), 256, 0, stream>>>(
      feat, Wc, fc7, Rroi, 0, nullptr, nullptr);
  // GEMM2: (PB @ M) .* PA, reduced per roi   (6272 x 1024 x 2048)
  wmma_gemm_kernel<<<dim3(Dd / BN, NCOLS / BM), 256, 0, stream>>>(
      PB, Mb, nullptr, NCOLS, 1, PA, scoreP);
  bbox_kernel<<<Rroi, 256, 0, stream>>>(fc7, bc, Wb, bb, out);
  score_final_kernel<<<1, 256, 0, stream>>>(scoreP, bs, out + (size_t)Rroi * 4);
}